// FlexAttention_6700148982216
// MI455X (gfx1250) — compile-verified
//
#include <hip/hip_runtime.h>

// ---------------------------------------------------------------------------
// Paged-attention decode (GQA 32q/8kv, D=128, page=16) for gfx1250 (MI455X).
// Bandwidth-bound (~512 MB KV -> ~22us @ 23.3 TB/s). Uses f32 WMMA
// V_WMMA_F32_16X16X4_F32 for QK^T and PV, flash-decoding online softmax,
// 8-wave split-K per (seq, kv_head) with LDS combine. wave32 throughout.
// ---------------------------------------------------------------------------

#define NUM_HEADS      32
#define HEAD_DIM       128
#define NUM_KV_HEADS   8
#define GQA            4          // query heads per kv head
#define BATCH          32
#define BLOCK_SIZE     16
#define BLOCKS_PER_SEQ 128
#define KV_ROW_STRIDE  (NUM_KV_HEADS * HEAD_DIM)   // floats between keys
#define SCALE          0.08838834764831845f
#define WAVES          8
#define THREADS        (WAVES * 32)
#define NEG_BIG        (-1.0e30f)

typedef float v2f __attribute__((ext_vector_type(2)));
typedef float v8f __attribute__((ext_vector_type(8)));

__device__ __forceinline__ float shflx16(float x) {
  return __shfl_xor(x, 16, 32);
}

__global__ __launch_bounds__(THREADS, 1)
void pa_decode_wmma(const float* __restrict__ qg,
                    const float* __restrict__ kc,
                    const float* __restrict__ vc,
                    const int*   __restrict__ bt,
                    const int*   __restrict__ cl,
                    float*       __restrict__ out)
{
  const int kvh   = blockIdx.x;       // 0..7
  const int sid   = blockIdx.y;       // 0..31
  const int tid   = threadIdx.x;
  const int wave  = tid >> 5;
  const int lane  = tid & 31;
  const int row16 = lane & 15;
  const int hi    = lane >> 4;        // 0 / 1 (lane half)
  const int h2    = hi << 1;          // 0 / 2 (K-dim sub-offset per ISA layout)

  const int  ctx    = cl[sid];
  const int  npages = (ctx + BLOCK_SIZE - 1) / BLOCK_SIZE;
  const int* btrow  = bt + sid * BLOCKS_PER_SEQ;

  // ---- Q staged once in B-operand layout: chunk c covers dims 4c..4c+3.
  // B 4x16 layout: VGPR0 = rows {K=0 | K=2}, VGPR1 = rows {K=1 | K=3},
  // lane n (mod 16) = column n = query head g (only g<4 live, rest zero).
  v2f qb[32];
  {
    const float* qp = qg + (size_t)sid * (NUM_HEADS * HEAD_DIM)
                         + (size_t)(kvh * GQA) * HEAD_DIM;
    #pragma unroll
    for (int c = 0; c < 32; ++c) {
      v2f t = {0.0f, 0.0f};
      if (row16 < GQA)
        t = *(const v2f*)(qp + row16 * HEAD_DIM + c * 4 + h2);   // 8B aligned
      qb[c] = t;
    }
  }

  // Output accumulator: D2[q=M][d=N], 8 column-chunks of 16.
  v8f acc[8];
  #pragma unroll
  for (int t = 0; t < 8; ++t)
    acc[t] = (v8f){0.f,0.f,0.f,0.f,0.f,0.f,0.f,0.f};
  float m_run = NEG_BIG, l_run = 0.0f;

  for (int p = wave; p < npages; p += WAVES) {
    const int blk = btrow[p];                               // scalar (uniform)
    const float* kpage = kc + (size_t)blk * (BLOCK_SIZE * KV_ROW_STRIDE)
                            + (size_t)kvh * HEAD_DIM;
    const float* vpage = vc + (size_t)blk * (BLOCK_SIZE * KV_ROW_STRIDE)
                            + (size_t)kvh * HEAD_DIM;

    // Prefetch next owned page (global_prefetch_b8).
    if (p + WAVES < npages) {
      const int nblk = btrow[p + WAVES];
      __builtin_prefetch(kc + (size_t)nblk * (BLOCK_SIZE * KV_ROW_STRIDE)
                            + (size_t)kvh * HEAD_DIM + lane * 16, 0, 1);
      __builtin_prefetch(vc + (size_t)nblk * (BLOCK_SIZE * KV_ROW_STRIDE)
                            + (size_t)kvh * HEAD_DIM + lane * 16, 0, 1);
    }

    // ---- QK^T: st[16 keys x 16 queries] = sum_c K_chunk(16x4) * Q_chunk(4x16)
    // A 16x4 layout: lane m<16 -> K[key m][d0..d0+1]; lane m+16 -> K[key m][d0+2..3]
    v8f st = (v8f){0.f,0.f,0.f,0.f,0.f,0.f,0.f,0.f};
    #pragma unroll
    for (int c = 0; c < 32; ++c) {
      v2f a = *(const v2f*)(kpage + (size_t)row16 * KV_ROW_STRIDE + c * 4 + h2);
      st = __builtin_amdgcn_wmma_f32_16x16x4_f32(false, a, false, qb[c],
                                                 (short)0, st, false, false);
    }

    // ---- scale + validity mask.  D layout: st[r] @ lane l = score for
    // key (p*16 + r + 8*hi), query (l & 15).
    float s[8];
    const int kbase = p * BLOCK_SIZE + (hi << 3);
    #pragma unroll
    for (int r = 0; r < 8; ++r) {
      float v = st[r] * SCALE;
      s[r] = (kbase + r < ctx) ? v : NEG_BIG;
    }

    // ---- online softmax (per query: 8 regs in-lane + xor16 lane pair)
    float mloc = s[0];
    #pragma unroll
    for (int r = 1; r < 8; ++r) mloc = fmaxf(mloc, s[r]);
    mloc = fmaxf(mloc, shflx16(mloc));
    const float m_new = fmaxf(m_run, mloc);
    const float alpha = __expf(m_run - m_new);
    float pt[8], rs = 0.0f;
    #pragma unroll
    for (int r = 0; r < 8; ++r) { pt[r] = __expf(s[r] - m_new); rs += pt[r]; }
    rs += shflx16(rs);
    l_run = l_run * alpha + rs;
    m_run = m_new;

    // rescale live accumulator rows (q = 0..3); alpha for query q lives @ lane q
    const float a0 = __shfl(alpha, 0, 32);
    const float a1 = __shfl(alpha, 1, 32);
    const float a2 = __shfl(alpha, 2, 32);
    const float a3 = __shfl(alpha, 3, 32);
    #pragma unroll
    for (int t = 0; t < 8; ++t) {
      acc[t][0] *= a0; acc[t][1] *= a1; acc[t][2] *= a2; acc[t][3] *= a3;
    }

    // ---- PV: acc[q][d] += sum_j P_chunk(16x4) * V_chunk(4x16)
    // Build A2 chunk j from transposed-P tile (pt[k&7] @ lane q + 16*(k>=8)).
    #pragma unroll
    for (int j = 0; j < 4; ++j) {
      const int rlo = (4 * j) & 7;        // 0,4,0,4
      const int rhi = (4 * j + 2) & 7;    // 2,6,2,6
      const float shi0 = shflx16(pt[rhi]);
      const float shi1 = shflx16(pt[rhi + 1]);
      const float slo0 = shflx16(pt[rlo]);
      const float slo1 = shflx16(pt[rlo + 1]);
      v2f a2v;
      if (j < 2) {   // k<8: low lanes self, high lanes from partner
        a2v.x = (hi == 0) ? pt[rlo]     : shi0;
        a2v.y = (hi == 0) ? pt[rlo + 1] : shi1;
      } else {       // k>=8: low lanes from partner, high lanes self
        a2v.x = (hi == 0) ? slo0 : pt[rhi];
        a2v.y = (hi == 0) ? slo1 : pt[rhi + 1];
      }
      // B 4x16 layout: .x = V[4j+h2][d], .y = V[4j+h2+1][d], d = 16t + (lane&15)
      #pragma unroll
      for (int t = 0; t < 8; ++t) {
        const float* vp = vpage + (size_t)(4 * j + h2) * KV_ROW_STRIDE
                                + t * 16 + row16;
        v2f b;
        b.x = vp[0];
        b.y = vp[KV_ROW_STRIDE];
        acc[t] = __builtin_amdgcn_wmma_f32_16x16x4_f32(false, a2v, false, b,
                                                       (short)0, acc[t],
                                                       false, false);
      }
    }
  }

  // ---- cross-wave combine (flash-decoding merge) ----
  __shared__ float s_acc[WAVES][GQA][HEAD_DIM];
  __shared__ float s_m[WAVES][GQA];
  __shared__ float s_l[WAVES][GQA];

  if (hi == 0) {
    #pragma unroll
    for (int t = 0; t < 8; ++t)
      #pragma unroll
      for (int r = 0; r < GQA; ++r)
        s_acc[wave][r][t * 16 + row16] = acc[t][r];
  }
  if (lane < GQA) { s_m[wave][lane] = m_run; s_l[wave][lane] = l_run; }
  __syncthreads();

  // 512 outputs over 256 threads (2 each)
  for (int e = tid; e < GQA * HEAD_DIM; e += THREADS) {
    const int qi = e >> 7;         // query head in group
    const int d  = e & 127;
    float M = NEG_BIG;
    #pragma unroll
    for (int w = 0; w < WAVES; ++w) M = fmaxf(M, s_m[w][qi]);
    float L = 0.0f, O = 0.0f;
    #pragma unroll
    for (int w = 0; w < WAVES; ++w) {
      const float f = __expf(s_m[w][qi] - M);
      L += f * s_l[w][qi];
      O += f * s_acc[w][qi][d];
    }
    out[(size_t)sid * (NUM_HEADS * HEAD_DIM)
        + (size_t)(kvh * GQA + qi) * HEAD_DIM + d] = O / L;
  }
}

extern "C" void kernel_launch(void* const* d_in, const int* in_sizes, int n_in,
                              void* d_out, int out_size, void* d_ws, size_t ws_size,
                              hipStream_t stream) {
  (void)in_sizes; (void)n_in; (void)out_size; (void)d_ws; (void)ws_size;
  const float* q  = (const float*)d_in[0];
  const float* kc = (const float*)d_in[1];
  const float* vc = (const float*)d_in[2];
  const int*   bt = (const int*)d_in[3];
  const int*   cl = (const int*)d_in[4];
  float*       o  = (float*)d_out;
  dim3 grid(NUM_KV_HEADS, BATCH);
  pa_decode_wmma<<<grid, THREADS, 0, stream>>>(q, kc, vc, bt, cl, o);
}